// PostProcessor_10325101379686
// MI455X (gfx1250) — compile-verified
//
#include <hip/hip_runtime.h>
#include <stdint.h>

#define N_FEAT     85
#define MAX_BOXES  1000
#define IOU_THR    0.5f
#define SCORE_THR  0.3f
#define NWG        75
#define TPB        512
#define NWAVES     (TPB / 32)
#define CHUNK_MAX  4096
#define NEG_INF_KEY 0x007FFFFFu   // fkey(-inf)

// Monotone order-preserving float -> uint key (IEEE total order, no NaNs here).
__device__ __forceinline__ unsigned fkey(float f) {
  unsigned u = __float_as_uint(f);
  return (u & 0x80000000u) ? ~u : (u | 0x80000000u);
}
// Packed argmax key: higher score wins; ties -> smaller index wins
// (matches jnp.argmax first-occurrence semantics).
__device__ __forceinline__ unsigned long long packKey(float s, unsigned idx) {
  return ((unsigned long long)fkey(s) << 32) |
         (unsigned long long)(0xFFFFFFFFu - idx);
}

// Monotonic-counter grid barrier (deterministic, no reset races).
__device__ __forceinline__ void grid_barrier(unsigned* cnt, unsigned b) {
  __syncthreads();
  if (threadIdx.x == 0) {
    __threadfence();
    atomicAdd(cnt, 1u);
    const unsigned want = (b + 1u) * (unsigned)NWG;
    while (atomicAdd(cnt, 0u) < want) { __builtin_amdgcn_s_sleep(1); }
    __threadfence();
  }
  __syncthreads();
}

// wave32 shuffle reduction + cross-wave LDS combine; result valid on tid==0.
__device__ __forceinline__ unsigned long long
block_max(unsigned long long v, unsigned long long* red) {
  const int tid = threadIdx.x;
#pragma unroll
  for (int off = 16; off > 0; off >>= 1) {
    unsigned long long o = __shfl_down(v, off, 32);
    if (o > v) v = o;
  }
  if ((tid & 31) == 0) red[tid >> 5] = v;
  __syncthreads();
  unsigned long long m = 0ull;
  if (tid == 0) {
    m = red[0];
#pragma unroll
    for (int i = 1; i < NWAVES; ++i)
      if (red[i] > m) m = red[i];
  }
  return m;
}

// ---------------- Phase 1: per-box class score / argmax ----------------
__global__ void score_kernel(const float* __restrict__ in, int n,
                             float4* __restrict__ bx,
                             float* __restrict__ clsf, float* __restrict__ clss,
                             float* __restrict__ msk) {
  int i = blockIdx.x * blockDim.x + threadIdx.x;
  if (i >= n) return;
  const float* row = in + (size_t)i * N_FEAT;
  float v0 = row[0], v1 = row[1], v2 = row[2], v3 = row[3];
  float conf = row[4];
  float best = -INFINITY; int bj = 0;
#pragma unroll 8
  for (int j = 0; j < N_FEAT - 5; ++j) {
    float v = conf * row[5 + j];
    if (v > best) { best = v; bj = j; }   // strict '>' keeps first max
  }
  bx[i] = make_float4(v0, v1, v2, v3);
  clsf[i] = (float)bj;
  clss[i] = best;
  msk[i] = (best >= SCORE_THR) ? best : -INFINITY;
}

// ---------------- Phase 2: persistent-grid greedy NMS ----------------
__global__ __launch_bounds__(TPB) void nms_kernel(
    const float4* __restrict__ bx, const float* __restrict__ msk,
    int n, int chunk,
    unsigned long long* __restrict__ slots,
    int* __restrict__ out_idx, int* __restrict__ out_valid,
    unsigned* __restrict__ barcnt) {
  __shared__ float4 sbox[CHUNK_MAX];          // boxes, LDS-resident (64KB)
  __shared__ float  sc[CHUNK_MAX];            // live masked scores (16KB)
  __shared__ unsigned long long red[NWAVES];  // cross-wave reduction scratch
  __shared__ float wbox[5];                   // winner y1,x1,y2,x2,area
  __shared__ unsigned winfo[2];               // winner idx, valid

  const int wg  = blockIdx.x;
  const int tid = threadIdx.x;
  const int base = wg * chunk;
  int lim = n - base;
  if (lim > chunk) lim = chunk;
  if (lim < 0) lim = 0;

  // ---- Stage boxes + scores into LDS with the CDNA5 async copy engine ----
  // Boxes: one GLOBAL_LOAD_ASYNC_TO_LDS_B128 per element (16B, aligned).
  for (int e = tid; e < lim; e += TPB) {
    unsigned lds_addr = (unsigned)(unsigned long long)(void*)&sbox[e];
    const float4* g = bx + (base + e);
    asm volatile("global_load_async_to_lds_b128 %0, %1, off"
                 :: "v"(lds_addr), "v"(g) : "memory");
  }
  // Scores: vectorized 16B async copies + scalar tail.
  const int lim4 = lim & ~3;
  for (int e = tid * 4; e < lim4; e += TPB * 4) {
    unsigned lds_addr = (unsigned)(unsigned long long)(void*)&sc[e];
    const float* g = msk + (base + e);
    asm volatile("global_load_async_to_lds_b128 %0, %1, off"
                 :: "v"(lds_addr), "v"(g) : "memory");
  }
  for (int e = lim4 + tid; e < lim; e += TPB) {
    unsigned lds_addr = (unsigned)(unsigned long long)(void*)&sc[e];
    const float* g = msk + (base + e);
    asm volatile("global_load_async_to_lds_b32 %0, %1, off"
                 :: "v"(lds_addr), "v"(g) : "memory");
  }
  asm volatile("s_wait_asynccnt 0x0" ::: "memory");
  __syncthreads();

  // ---- Initial argmax over this chunk -> slots[0] ----
  unsigned long long lm = 0ull;
  for (int e = tid; e < lim; e += TPB) {
    unsigned long long k = packKey(sc[e], (unsigned)(base + e));
    if (k > lm) lm = k;
  }
  {
    unsigned long long m = block_max(lm, red);
    if (tid == 0) atomicMax(&slots[0], m);
  }

  unsigned bar = 0;
  grid_barrier(barcnt, bar++);   // slots[0] complete everywhere

  for (int k = 0; k < MAX_BOXES; ++k) {
    if (tid == 0) {
      unsigned long long w = atomicAdd(&slots[k], 0ull);  // coherent read
      unsigned widx = 0xFFFFFFFFu - (unsigned)(w & 0xFFFFFFFFull);
      unsigned hkey = (unsigned)(w >> 32);
      unsigned wv = (hkey != NEG_INF_KEY) ? 1u : 0u;
      float4 wb = bx[widx];
      wbox[0] = wb.x; wbox[1] = wb.y; wbox[2] = wb.z; wbox[3] = wb.w;
      wbox[4] = (wb.z - wb.x) * (wb.w - wb.y);
      winfo[0] = widx; winfo[1] = wv;
      if (wg == 0) { out_idx[k] = (int)widx; out_valid[k] = (int)wv; }
    }
    __syncthreads();
    const unsigned widx = winfo[0];
    const unsigned wv   = winfo[1];
    const float wy1 = wbox[0], wx1 = wbox[1], wy2 = wbox[2], wx2 = wbox[3];
    const float warea = wbox[4];

    // Fused: suppress vs. winner AND compute next argmax — all from LDS.
    unsigned long long lmax = 0ull;
    for (int e = tid; e < lim; e += TPB) {
      const unsigned g = (unsigned)(base + e);
      float s = sc[e];
      if (g == widx) {
        s = -INFINITY;                       // scores.at[idx].set(-inf)
      } else if (wv && s != -INFINITY) {
        float4 b = sbox[e];
        float iy1 = fmaxf(wy1, b.x), ix1 = fmaxf(wx1, b.y);
        float iy2 = fminf(wy2, b.z), ix2 = fminf(wx2, b.w);
        float inter = fmaxf(iy2 - iy1, 0.0f) * fmaxf(ix2 - ix1, 0.0f);
        float uni = warea + (b.z - b.x) * (b.w - b.y) - inter;
        float iou = (uni > 0.0f) ? inter / uni : 0.0f;
        if (iou > IOU_THR) s = -INFINITY;
      }
      sc[e] = s;
      unsigned long long kk = packKey(s, g);
      if (kk > lmax) lmax = kk;
    }
    unsigned long long m = block_max(lmax, red);
    if (tid == 0) atomicMax(&slots[k + 1], m);
    grid_barrier(barcnt, bar++);
  }
}

// ---------------- Phase 3: gather outputs ----------------
__global__ void gather_kernel(const float4* __restrict__ bx,
                              const float* __restrict__ clsf,
                              const float* __restrict__ clss,
                              const int* __restrict__ out_idx,
                              const int* __restrict__ out_valid,
                              float* __restrict__ out) {
  int k = blockIdx.x * blockDim.x + threadIdx.x;
  if (k >= MAX_BOXES) return;
  int idx = out_idx[k];
  int v = out_valid[k];
  float* o = out + (size_t)k * 6;
  if (v) {
    float4 b = bx[idx];
    o[0] = b.x; o[1] = b.y; o[2] = b.z; o[3] = b.w;
    o[4] = clsf[idx]; o[5] = clss[idx];
  } else {
    o[0] = 0.f; o[1] = 0.f; o[2] = 0.f; o[3] = 0.f; o[4] = 0.f; o[5] = 0.f;
  }
}

extern "C" void kernel_launch(void* const* d_in, const int* in_sizes, int n_in,
                              void* d_out, int out_size, void* d_ws, size_t ws_size,
                              hipStream_t stream) {
  (void)n_in; (void)out_size; (void)ws_size;
  const float* in = (const float*)d_in[0];
  const int n = in_sizes[0] / N_FEAT;

  char* ws = (char*)d_ws;
  const size_t fN = (size_t)n * sizeof(float);
  float4* bx  = (float4*)(ws);              // 4N floats, 16B-aligned AoS
  float* clsf = (float*)(ws + 4 * fN);
  float* clss = (float*)(ws + 5 * fN);
  float* msk  = (float*)(ws + 6 * fN);

  const size_t ctrl_off = (7 * fN + 255) & ~(size_t)255;
  unsigned long long* slots = (unsigned long long*)(ws + ctrl_off);
  const size_t slots_bytes = (size_t)(MAX_BOXES + 8) * 8;
  int* out_idx   = (int*)(ws + ctrl_off + slots_bytes);
  int* out_valid = out_idx + MAX_BOXES;
  unsigned* barcnt = (unsigned*)(out_valid + MAX_BOXES);
  const size_t ctrl_bytes = slots_bytes + 2 * (size_t)MAX_BOXES * 4 + 64;

  // Zero argmax slots + barrier counter each call (graph-capture safe).
  hipMemsetAsync(ws + ctrl_off, 0, ctrl_bytes, stream);

  const int chunk = (n + NWG - 1) / NWG;   // 4000 for N=300000

  score_kernel<<<(n + 255) / 256, 256, 0, stream>>>(in, n, bx, clsf, clss, msk);
  nms_kernel<<<NWG, TPB, 0, stream>>>(bx, msk, n, chunk,
                                      slots, out_idx, out_valid, barcnt);
  gather_kernel<<<(MAX_BOXES + 255) / 256, 256, 0, stream>>>(
      bx, clsf, clss, out_idx, out_valid, (float*)d_out);
}